// GCNModel_45406394253339
// MI455X (gfx1250) — compile-verified
//
#include <hip/hip_runtime.h>

#define NN 100000
#define NE 3200000
#define DIM 512
#define LDS_STRIDE 516   // 512 + 4 pad (TDM inserts 2 dwords every 256 dwords)

typedef __attribute__((ext_vector_type(2))) float v2f;
typedef __attribute__((ext_vector_type(8))) float v8f;
typedef __attribute__((ext_vector_type(4))) unsigned int u32x4;
typedef __attribute__((ext_vector_type(8))) int i32x8;
typedef __attribute__((ext_vector_type(4))) int i32x4;

// ---------------- utility ----------------
__global__ void k_zero4(float4* p, int n4) {
    int i = blockIdx.x * blockDim.x + threadIdx.x;
    if (i < n4) p[i] = make_float4(0.f, 0.f, 0.f, 0.f);
}

// ---------------- degrees / norms ----------------
__global__ void k_deg_acc(const int* __restrict__ src, const int* __restrict__ dst,
                          int* deg_out, int* deg_in) {
    int e = blockIdx.x * blockDim.x + threadIdx.x;
    if (e < NE) {
        atomicAdd(&deg_out[src[e]], 1);
        atomicAdd(&deg_in[dst[e]], 1);
    }
}

__global__ void k_norms(const int* __restrict__ deg_out, const int* __restrict__ deg_in,
                        float* n_out, float* n_in) {
    int i = blockIdx.x * blockDim.x + threadIdx.x;
    if (i < NN) {
        n_out[i] = rsqrtf((float)deg_out[i] + 1.0f);
        n_in[i]  = rsqrtf((float)deg_in[i] + 1.0f);
    }
}

// ---------------- input feature lift: [w, emb0, emb1] @ lin_W + lin_b ----------------
__global__ void k_feats(const float* __restrict__ weight, const int* __restrict__ sig,
                        const float* __restrict__ emb, const float* __restrict__ linW,
                        const float* __restrict__ linb, float* __restrict__ out) {
    int tid = blockIdx.x * blockDim.x + threadIdx.x;      // N * DIM/4 threads
    if (tid >= NN * (DIM / 4)) return;
    int n  = tid / (DIM / 4);
    int c  = (tid % (DIM / 4)) * 4;
    float w = weight[n];
    int   s = sig[n];
    float e0 = emb[2 * s + 0], e1 = emb[2 * s + 1];
    float4 r;
    r.x = w * linW[c + 0] + e0 * linW[DIM + c + 0] + e1 * linW[2 * DIM + c + 0] + linb[c + 0];
    r.y = w * linW[c + 1] + e0 * linW[DIM + c + 1] + e1 * linW[2 * DIM + c + 1] + linb[c + 1];
    r.z = w * linW[c + 2] + e0 * linW[DIM + c + 2] + e1 * linW[2 * DIM + c + 2] + linb[c + 2];
    r.w = w * linW[c + 3] + e0 * linW[DIM + c + 3] + e1 * linW[2 * DIM + c + 3] + linb[c + 3];
    *(float4*)&out[(size_t)n * DIM + c] = r;
}

// ---------------- SpMM: agg[dst] += h[src] * n_out[src]  (wave32 per edge) ----------------
__global__ void k_spmm(const float* __restrict__ h, const float* __restrict__ n_out,
                       const int* __restrict__ src, const int* __restrict__ dst,
                       float* __restrict__ agg) {
    int gid  = blockIdx.x * blockDim.x + threadIdx.x;
    int wave = gid >> 5;          // gfx1250: wave32
    int lane = threadIdx.x & 31;
    if (wave >= NE) return;
    int s = src[wave], d = dst[wave];
    float sc = n_out[s];
    const float4* hs = (const float4*)(h + (size_t)s * DIM);
    float* ad = agg + (size_t)d * DIM;
#pragma unroll
    for (int j = 0; j < 4; ++j) {
        int f4 = lane + 32 * j;          // coalesced float4 per lane
        float4 v = hs[f4];
        int b = f4 * 4;
        atomicAdd(&ad[b + 0], v.x * sc);
        atomicAdd(&ad[b + 1], v.y * sc);
        atomicAdd(&ad[b + 2], v.z * sc);
        atomicAdd(&ad[b + 3], v.w * sc);
    }
}

// ---------------- agg = (agg + h*n_out) * n_in ----------------
__global__ void k_combine(float* __restrict__ agg, const float* __restrict__ h,
                          const float* __restrict__ n_out, const float* __restrict__ n_in) {
    int tid = blockIdx.x * blockDim.x + threadIdx.x;
    if (tid >= NN * (DIM / 4)) return;
    int n = tid / (DIM / 4);
    size_t o = (size_t)n * DIM + (tid % (DIM / 4)) * 4;
    float no = n_out[n], ni = n_in[n];
    float4 a  = *(float4*)&agg[o];
    float4 hv = *(const float4*)&h[o];
    a.x = (a.x + hv.x * no) * ni;
    a.y = (a.y + hv.y * no) * ni;
    a.z = (a.z + hv.z * no) * ni;
    a.w = (a.w + hv.w * no) * ni;
    *(float4*)&agg[o] = a;
}

// ---------------- fp32 WMMA GEMM: out[16-row strip] = A @ W + b (+LeakyReLU) ----------------
// A strip (16x512 fp32 = 32KB) is DMA'd into LDS by the Tensor Data Mover with
// hardware padding (2 dwords per 256 dwords => effective row stride 516 floats,
// bank-conflict-free b64 A-fragment reads).
__global__ __launch_bounds__(256) void k_gemm(const float* __restrict__ A,
                                              const float* __restrict__ W,
                                              const float* __restrict__ bias,
                                              float* __restrict__ out, int leaky) {
    __shared__ __align__(16) float As[16 * LDS_STRIDE];
    int row0 = blockIdx.x * 16;

    if ((threadIdx.x >> 5) == 0) {   // wave 0 issues the TDM descriptor
        unsigned long long gaddr =
            (unsigned long long)(uintptr_t)(const void*)(A + (size_t)row0 * DIM);
        unsigned int lds = (unsigned int)(uintptr_t)(void*)As;  // LDS byte address (low 32 bits)
        u32x4 g0;
        g0[0] = 1u;                                   // count=1 (valid descriptor)
        g0[1] = lds;                                  // lds_addr (bytes)
        g0[2] = (unsigned int)gaddr;                  // global_addr[31:0]
        g0[3] = (unsigned int)((gaddr >> 32) & 0x01FFFFFFull) | (2u << 30); // hi bits + type=2
        i32x8 g1;
        g1[0] = (2 << 16)      // data_size = 4 bytes
              | (1 << 20)      // pad_enable
              | (7 << 22)      // pad_interval: every 256 dwords
              | (1 << 25);     // pad_amount: 2 dwords
        g1[1] = (int)(512u << 16);   // tensor_dim0[15:0] in bits[63:48]
        g1[2] = (int)(16u << 16);    // tensor_dim0 hi=0 | tensor_dim1[15:0]
        g1[3] = (int)(512u << 16);   // tensor_dim1 hi=0 | tile_dim0=512
        g1[4] = 16;                  // tile_dim1=16, tile_dim2=0
        g1[5] = 512;                 // tensor_dim0_stride lo
        g1[6] = 0;
        g1[7] = 0;
        i32x4 gz4 = {0, 0, 0, 0};
        i32x8 gz8 = {0, 0, 0, 0, 0, 0, 0, 0};
        __builtin_amdgcn_tensor_load_to_lds(g0, g1, gz4, gz4, gz8, 0);
        __builtin_amdgcn_s_wait_tensorcnt(0);
    }
    __syncthreads();

    int wave = threadIdx.x >> 5;
    int lane = threadIdx.x & 31;
    int half = lane >> 4;        // K-group select (ISA 16x4 f32 A/B layout)
    int l16  = lane & 15;
    const float* Arow = &As[l16 * LDS_STRIDE];

    for (int t = 0; t < 4; ++t) {
        int col0 = (wave + 8 * t) * 16;
        const float* Wc = W + col0 + l16;
        __builtin_prefetch(Wc, 0, 1);
        v8f c = {0.f, 0.f, 0.f, 0.f, 0.f, 0.f, 0.f, 0.f};
        for (int k = 0; k < DIM; k += 4) {
            int pk = k + ((k >= 256) ? 2 : 0);   // TDM pad lands at dword 256 of each row
            // A frag: lane holds A[M=l16][k+2*half .. +1]
            v2f a = *(const v2f*)&Arow[pk + 2 * half];
            // B frag: lane holds W[k+2*half .. +1][col0+l16]
            v2f b;
            b.x = Wc[(size_t)(k + 2 * half) * DIM];
            b.y = Wc[(size_t)(k + 2 * half + 1) * DIM];
            c = __builtin_amdgcn_wmma_f32_16x16x4_f32(false, a, false, b,
                                                      (short)0, c, false, false);
        }
        float bc = bias[col0 + l16];
#pragma unroll
        for (int r = 0; r < 8; ++r) {
            int row = row0 + r + 8 * half;   // C/D layout: M = r + 8*(lane>=16)
            float v = c[r] + bc;
            if (leaky) v = (v >= 0.f) ? v : 0.01f * v;
            out[(size_t)row * DIM + col0 + l16] = v;
        }
    }
}

// ---------------- prediction head: out[n] = h[n,:] . pred_W + pred_b ----------------
__global__ void k_pred(const float* __restrict__ h, const float* __restrict__ pw,
                       const float* __restrict__ pb, float* __restrict__ out) {
    int gid  = blockIdx.x * blockDim.x + threadIdx.x;
    int wave = gid >> 5;
    int lane = threadIdx.x & 31;
    if (wave >= NN) return;
    const float4* hp = (const float4*)(h + (size_t)wave * DIM);
    const float4* wp = (const float4*)pw;
    float s = 0.f;
#pragma unroll
    for (int j = 0; j < 4; ++j) {
        float4 a = hp[lane + 32 * j];
        float4 b = wp[lane + 32 * j];
        s += a.x * b.x + a.y * b.y + a.z * b.z + a.w * b.w;
    }
    for (int off = 16; off > 0; off >>= 1) s += __shfl_down(s, off, 32);
    if (lane == 0) out[wave] = s + pb[0];
}

// ---------------- driver ----------------
extern "C" void kernel_launch(void* const* d_in, const int* in_sizes, int n_in,
                              void* d_out, int out_size, void* d_ws, size_t ws_size,
                              hipStream_t stream) {
    const float* weight = (const float*)d_in[0];
    const int*   sig    = (const int*)d_in[1];
    const int*   src    = (const int*)d_in[2];
    const int*   dst    = (const int*)d_in[3];
    const float* emb    = (const float*)d_in[4];
    const float* linW   = (const float*)d_in[5];
    const float* linb   = (const float*)d_in[6];
    const float* Wl[3]  = {(const float*)d_in[7], (const float*)d_in[9],  (const float*)d_in[11]};
    const float* bl[3]  = {(const float*)d_in[8], (const float*)d_in[10], (const float*)d_in[12]};
    const float* predW  = (const float*)d_in[13];
    const float* predB  = (const float*)d_in[14];
    float* outp = (float*)d_out;

    char* ws = (char*)d_ws;
    size_t off = 0;
    auto alloc = [&](size_t bytes) -> void* {
        off = (off + 255) & ~(size_t)255;
        void* p = ws + off;
        off += bytes;
        return p;
    };
    int*   degO = (int*)alloc((size_t)NN * 4);
    int*   degI = (int*)alloc((size_t)NN * 4);
    float* nO   = (float*)alloc((size_t)NN * 4);
    float* nI   = (float*)alloc((size_t)NN * 4);
    float* bufH = (float*)alloc((size_t)NN * DIM * 4);
    float* agg  = (float*)alloc((size_t)NN * DIM * 4);

    const int TB = 256;
    const int nd4      = NN * (DIM / 4);          // 12.8M
    const int g_nd4    = (nd4 + TB - 1) / TB;     // 50000
    const int g_deg0   = (2 * NN * 4 / 16 + TB - 1) / TB;
    const int g_edges  = (NE + TB - 1) / TB;      // 12500
    const int g_nodes  = (NN + TB - 1) / TB;      // 391
    const int g_spmm   = (NE * 32 + TB - 1) / TB; // 400000
    const int g_gemm   = NN / 16;                 // 6250 (exact)
    const int g_pred   = (NN * 32 + TB - 1) / TB; // 12500

    // degrees + norms
    k_zero4<<<g_deg0, TB, 0, stream>>>((float4*)degO, 2 * NN * 4 / 16);
    k_deg_acc<<<g_edges, TB, 0, stream>>>(src, dst, degO, degI);
    k_norms<<<g_nodes, TB, 0, stream>>>(degO, degI, nO, nI);

    // input features -> bufH
    k_feats<<<g_nd4, TB, 0, stream>>>(weight, sig, emb, linW, linb, bufH);

    // 3 GCN layers; LeakyReLU fused into epilogue of layers 0 and 1
    for (int l = 0; l < 3; ++l) {
        k_zero4<<<g_nd4, TB, 0, stream>>>((float4*)agg, nd4);
        k_spmm<<<g_spmm, TB, 0, stream>>>(bufH, nO, src, dst, agg);
        k_combine<<<g_nd4, TB, 0, stream>>>(agg, bufH, nO, nI);
        k_gemm<<<g_gemm, TB, 0, stream>>>(agg, Wl[l], bl[l], bufH, (l < 2) ? 1 : 0);
    }

    // prediction logits
    k_pred<<<g_pred, TB, 0, stream>>>(bufH, predW, predB, outp);
}